// GraphConvNetwork_31963146617555
// MI455X (gfx1250) — compile-verified
//
#include <hip/hip_runtime.h>
#include <hip/hip_bf16.h>

// ---------------------------------------------------------------------------
// GCN 2-layer forward on gfx1250 (MI455X).
//   layer: h = x @ W;  agg = h*dinv^2 + sum_{e:(s->d)} h[s]*dinv[s]*dinv[d]
//   then  +b, BN (eval), [ReLU]
// GEMMs: V_WMMA_F32_16X16X4_F32 (exact fp32), self-loop term fused into the
// WMMA epilogue. Aggregation: non-returning agent-scope fp32 atomics.
// ---------------------------------------------------------------------------

#define GCN_N_FEAT_IN  64
#define GCN_N_FEAT_HID 128
#define GCN_N_FEAT_OUT 64
#define GCN_BN_EPS 1e-5f

typedef __attribute__((ext_vector_type(2))) float v2f;
typedef __attribute__((ext_vector_type(8))) float v8f;

__device__ __forceinline__ void atomic_add_f32_agent(float* p, float v) {
    // Non-returning, relaxed, device-scope fp32 atomic -> global_atomic_add_f32
    (void)__hip_atomic_fetch_add(p, v, __ATOMIC_RELAXED, __HIP_MEMORY_SCOPE_AGENT);
}

// ---------------- degree / dinv ----------------
__global__ void gcn_deg_init(float* __restrict__ deg, int n) {
    int i = blockIdx.x * blockDim.x + threadIdx.x;
    if (i < n) deg[i] = 1.0f;  // +1 self loop
}

__global__ void gcn_deg_count(float* __restrict__ deg,
                              const long long* __restrict__ dst, int e) {
    int i = blockIdx.x * blockDim.x + threadIdx.x;
    if (i < e) atomic_add_f32_agent(&deg[(int)dst[i]], 1.0f);
}

__global__ void gcn_dinv(float* __restrict__ deg, int n) {
    int i = blockIdx.x * blockDim.x + threadIdx.x;
    if (i < n) deg[i] = rsqrtf(deg[i]);
}

// ---------------- WMMA fp32 GEMM + fused self-loop epilogue ----------------
// C[M,NC] = A[M,K] @ W[K,NC];  AGG[M,NC] = C * dinv[M]^2
// One block = 16 rows of A; one wave per 16-wide column tile (NC/16 waves).
// A tile and full W staged in LDS. All dims multiples of 16; EXEC all ones.
template <int K, int NC>
__global__ __launch_bounds__(32 * (NC / 16))
void gcn_gemm_wmma(const float* __restrict__ A, const float* __restrict__ W,
                   float* __restrict__ C, float* __restrict__ AGG,
                   const float* __restrict__ dinv) {
    __shared__ float sW[K * NC];
    __shared__ float sA[16][K + 1];  // +1 pad: avoid 16-way LDS bank conflicts

    const int tid  = threadIdx.x;
    const int bd   = 32 * (NC / 16);
    const int row0 = blockIdx.x * 16;

    for (int i = tid; i < K * NC; i += bd) sW[i] = W[i];
    for (int i = tid; i < 16 * K; i += bd) sA[i / K][i % K] = A[row0 * K + i];
    __syncthreads();

    const int wave = tid >> 5;
    const int lane = tid & 31;
    const int col0 = wave * 16;
    const int mn   = lane & 15;         // M index (A) / N index (B)
    const int kh   = (lane >> 4) * 2;   // K sub-offset per ISA fp32 layout

    v8f c = {};
#pragma unroll
    for (int k = 0; k < K; k += 4) {
        // A 16x4 fragment: lane L -> M=L&15; VGPR0: K=k+kh, VGPR1: K=k+kh+1
        v2f a;
        a.x = sA[mn][k + kh];
        a.y = sA[mn][k + kh + 1];
        // B 4x16 fragment: lane L -> N=L&15; VGPR0: K=k+kh, VGPR1: K=k+kh+1
        v2f b;
        b.x = sW[(k + kh)     * NC + col0 + mn];
        b.y = sW[(k + kh + 1) * NC + col0 + mn];
        c = __builtin_amdgcn_wmma_f32_16x16x4_f32(
                /*neg_a=*/false, a, /*neg_b=*/false, b,
                /*c_mod=*/(short)0, c, /*reuse_a=*/false, /*reuse_b=*/false);
    }

    // C/D 16x16 layout: VGPR r holds M = r + 8*(lane>>4), N = lane&15
#pragma unroll
    for (int r = 0; r < 8; ++r) {
        int m = r + (lane >> 4) * 8;
        float di = dinv[row0 + m];
        long  o  = (long)(row0 + m) * NC + col0 + mn;
        C[o]   = c[r];
        AGG[o] = c[r] * di * di;   // fused self-loop message
    }
}

// ---------------- edge scatter: agg[dst] += h[src] * dinv[s]*dinv[d] --------
// One thread handles 4 features of one edge (float4 gather, 4 atomics).
template <int F>
__global__ void gcn_scatter(float* __restrict__ agg, const float* __restrict__ h,
                            const float* __restrict__ dinv,
                            const long long* __restrict__ src,
                            const long long* __restrict__ dst, int e) {
    constexpr int QPE = F / 4;  // float4 quads per edge
    long idx = (long)blockIdx.x * blockDim.x + threadIdx.x;
    if (idx >= (long)e * QPE) return;
    int eid = (int)(idx / QPE);
    int q   = (int)(idx % QPE);
    int s = (int)src[eid];
    int d = (int)dst[eid];
    float coef = dinv[s] * dinv[d];
    const float4 hv = *(const float4*)(h + (long)s * F + q * 4);
    float* out = agg + (long)d * F + q * 4;
    atomic_add_f32_agent(out + 0, hv.x * coef);
    atomic_add_f32_agent(out + 1, hv.y * coef);
    atomic_add_f32_agent(out + 2, hv.z * coef);
    atomic_add_f32_agent(out + 3, hv.w * coef);
}

// ---------------- bias + BN (+ optional ReLU), in place, float4 -------------
template <int F, bool RELU>
__global__ void gcn_bias_bn(float* __restrict__ a, const float* __restrict__ b,
                            const float* __restrict__ g, const float* __restrict__ beta,
                            const float* __restrict__ m, const float* __restrict__ v,
                            int n) {
    constexpr int Q = F / 4;
    long idx = (long)blockIdx.x * blockDim.x + threadIdx.x;
    if (idx >= (long)n * Q) return;
    int q = (int)(idx % Q);
    float4 xv = *(float4*)(a + idx * 4);
    const float4 bv  = *(const float4*)(b    + q * 4);
    const float4 gv  = *(const float4*)(g    + q * 4);
    const float4 be  = *(const float4*)(beta + q * 4);
    const float4 mv  = *(const float4*)(m    + q * 4);
    const float4 vv  = *(const float4*)(v    + q * 4);
    float4 y;
    y.x = gv.x * (xv.x + bv.x - mv.x) * rsqrtf(vv.x + GCN_BN_EPS) + be.x;
    y.y = gv.y * (xv.y + bv.y - mv.y) * rsqrtf(vv.y + GCN_BN_EPS) + be.y;
    y.z = gv.z * (xv.z + bv.z - mv.z) * rsqrtf(vv.z + GCN_BN_EPS) + be.z;
    y.w = gv.w * (xv.w + bv.w - mv.w) * rsqrtf(vv.w + GCN_BN_EPS) + be.w;
    if (RELU) {
        y.x = fmaxf(y.x, 0.0f); y.y = fmaxf(y.y, 0.0f);
        y.z = fmaxf(y.z, 0.0f); y.w = fmaxf(y.w, 0.0f);
    }
    *(float4*)(a + idx * 4) = y;
}

// ---------------------------------------------------------------------------
extern "C" void kernel_launch(void* const* d_in, const int* in_sizes, int n_in,
                              void* d_out, int out_size, void* d_ws, size_t ws_size,
                              hipStream_t stream) {
    const float*     x    = (const float*)d_in[0];
    const long long* ei   = (const long long*)d_in[1];  // int64 [2, E]
    const float*     W1   = (const float*)d_in[2];
    const float*     b1   = (const float*)d_in[3];
    const float*     g1   = (const float*)d_in[4];
    const float*     bt1  = (const float*)d_in[5];
    const float*     m1   = (const float*)d_in[6];
    const float*     v1   = (const float*)d_in[7];
    const float*     W2   = (const float*)d_in[8];
    const float*     b2   = (const float*)d_in[9];
    const float*     g2   = (const float*)d_in[10];
    const float*     bt2  = (const float*)d_in[11];
    const float*     m2   = (const float*)d_in[12];
    const float*     v2   = (const float*)d_in[13];

    const int N = in_sizes[0] / GCN_N_FEAT_IN;
    const int E = in_sizes[1] / 2;
    const long long* srcp = ei;
    const long long* dstp = ei + E;

    // workspace: dinv[N] | buf1[N*128] (h1 then h2) | buf2[N*128] (agg1/hbn)
    float* dinv = (float*)d_ws;
    float* h1   = dinv + N;
    float* agg1 = h1 + (long)N * GCN_N_FEAT_HID;
    float* h2   = h1;              // h1 dead after layer-1 BN
    float* outf = (float*)d_out;   // agg2 accumulates directly in d_out

    const int T = 256;
    // 1) degrees -> dinv
    gcn_deg_init<<<(N + T - 1) / T, T, 0, stream>>>(dinv, N);
    gcn_deg_count<<<(E + T - 1) / T, T, 0, stream>>>(dinv, dstp, E);
    gcn_dinv<<<(N + T - 1) / T, T, 0, stream>>>(dinv, N);

    // 2) h1 = x @ W1, agg1 = h1*dinv^2 fused   [N,64]@[64,128]
    gcn_gemm_wmma<GCN_N_FEAT_IN, GCN_N_FEAT_HID>
        <<<N / 16, 32 * (GCN_N_FEAT_HID / 16), 0, stream>>>(x, W1, h1, agg1, dinv);

    // 3) aggregate layer 1 + bias/BN/ReLU
    {
        long w = (long)E * (GCN_N_FEAT_HID / 4);
        gcn_scatter<GCN_N_FEAT_HID><<<(w + T - 1) / T, T, 0, stream>>>(agg1, h1, dinv, srcp, dstp, E);
        long nq = (long)N * (GCN_N_FEAT_HID / 4);
        gcn_bias_bn<GCN_N_FEAT_HID, true><<<(nq + T - 1) / T, T, 0, stream>>>(agg1, b1, g1, bt1, m1, v1, N);
    }

    // 4) h2 = hbn @ W2, out = h2*dinv^2 fused  [N,128]@[128,64]
    gcn_gemm_wmma<GCN_N_FEAT_HID, GCN_N_FEAT_OUT>
        <<<N / 16, 32 * (GCN_N_FEAT_OUT / 16), 0, stream>>>(agg1, W2, h2, outf, dinv);

    // 5) aggregate layer 2 into d_out, final bias/BN
    {
        long w = (long)E * (GCN_N_FEAT_OUT / 4);
        gcn_scatter<GCN_N_FEAT_OUT><<<(w + T - 1) / T, T, 0, stream>>>(outf, h2, dinv, srcp, dstp, E);
        long nq = (long)N * (GCN_N_FEAT_OUT / 4);
        gcn_bias_bn<GCN_N_FEAT_OUT, false><<<(nq + T - 1) / T, T, 0, stream>>>(outf, b2, g2, bt2, m2, v2, N);
    }
}